// HRLM_68006512165286
// MI455X (gfx1250) — compile-verified
//
#include <hip/hip_runtime.h>
#include <math.h>

// ---------------- Types for WMMA ----------------
typedef __attribute__((ext_vector_type(2))) float v2f;
typedef __attribute__((ext_vector_type(8))) float v8f;

static __device__ inline v8f wmma4(v2f a, v2f b, v8f c) {
  // D = A(16x4 f32) * B(4x16 f32) + C(16x16 f32)
  return __builtin_amdgcn_wmma_f32_16x16x4_f32(
      /*neg_a=*/false, a, /*neg_b=*/false, b,
      /*c_mod=*/(short)0, c, /*reuse_a=*/false, /*reuse_b=*/false);
}

static __device__ inline float warp_sum(float v) {
  for (int off = 16; off >= 1; off >>= 1) v += __shfl_xor(v, off, 32);
  return v;
}
static __device__ inline float warp_max(float v) {
  for (int off = 16; off >= 1; off >>= 1) v = fmaxf(v, __shfl_xor(v, off, 32));
  return v;
}

// Constants
#define BB 8
#define CC 512
#define NN 256      // 16x16
#define IMG (CC*NN) // per-batch image floats = 131072
#define BIGF (BB*IMG)

// ---------------- Row mean (GAP): one wave per row ----------------
__global__ __launch_bounds__(32) void row_mean_kernel(const float* __restrict__ x,
                                                      float* __restrict__ out, int len) {
  int row = blockIdx.x;
  int lane = threadIdx.x;
  const float* p = x + (size_t)row * len;
  float s = 0.f;
  for (int i = lane; i < len; i += 32) s += p[i];
  s = warp_sum(s);
  if (lane == 0) out[row] = s / (float)len;
}

// ---------------- Small GEMM: out[M,N] = act(A[M,K] @ W[N,K]^T + bias) (+res) ----
__global__ void small_gemm_kernel(const float* __restrict__ A, const float* __restrict__ W,
                                  const float* __restrict__ bias, const float* __restrict__ res,
                                  float* __restrict__ out, int M, int N, int K, int relu) {
  int idx = blockIdx.x * blockDim.x + threadIdx.x;
  if (idx >= M * N) return;
  int m = idx / N, n = idx % N;
  const float* a = A + (size_t)m * K;
  const float* w = W + (size_t)n * K;
  float acc = bias ? bias[n] : 0.f;
  for (int k = 0; k < K; k++) acc = fmaf(a[k], w[k], acc);
  if (relu) acc = fmaxf(acc, 0.f);
  if (res) acc += res[idx];
  out[idx] = acc;
}

// ---------------- LayerNorm over contiguous rows of length len ----------------
__global__ void ln_row_kernel(const float* __restrict__ x, const float* __restrict__ g,
                              const float* __restrict__ b, float* __restrict__ out, int len) {
  int row = blockIdx.x;
  int t = threadIdx.x;                 // blockDim = 256
  const float* p = x + (size_t)row * len;
  float s = 0.f, s2 = 0.f;
  for (int i = t; i < len; i += 256) { float v = p[i]; s += v; s2 += v * v; }
  s = warp_sum(s); s2 = warp_sum(s2);
  __shared__ float w1[8], w2[8];
  int wid = t >> 5, lid = t & 31;
  if (lid == 0) { w1[wid] = s; w2[wid] = s2; }
  __syncthreads();
  if (t == 0) {
    float a = 0.f, c = 0.f;
    for (int i = 0; i < 8; i++) { a += w1[i]; c += w2[i]; }
    w1[0] = a; w2[0] = c;
  }
  __syncthreads();
  float mean = w1[0] / (float)len;
  float var  = w2[0] / (float)len - mean * mean;
  float inv  = rsqrtf(var + 1e-5f);
  for (int i = t; i < len; i += 256)
    out[(size_t)row * len + i] = (p[i] - mean) * inv * g[i] + b[i];
}

// ---------------- LayerNorm over channel dim for [B,C,N] layout (per (b,n)) -----
__global__ void ln_chan_kernel(const float* __restrict__ x, const float* __restrict__ g,
                               const float* __restrict__ bta, float* __restrict__ out) {
  int r = blockIdx.x;          // 0..2047 : b*256 + n
  int b = r >> 8, n = r & 255;
  int t = threadIdx.x;         // 256
  const float* base = x + (size_t)b * IMG + n;
  float s = 0.f, s2 = 0.f;
  for (int c = t; c < CC; c += 256) { float v = base[(size_t)c * NN]; s += v; s2 += v * v; }
  s = warp_sum(s); s2 = warp_sum(s2);
  __shared__ float w1[8], w2[8];
  int wid = t >> 5, lid = t & 31;
  if (lid == 0) { w1[wid] = s; w2[wid] = s2; }
  __syncthreads();
  if (t == 0) {
    float a = 0.f, c2 = 0.f;
    for (int i = 0; i < 8; i++) { a += w1[i]; c2 += w2[i]; }
    w1[0] = a; w2[0] = c2;
  }
  __syncthreads();
  float mean = w1[0] / (float)CC;
  float var  = w2[0] / (float)CC - mean * mean;
  float inv  = rsqrtf(var + 1e-5f);
  for (int c = t; c < CC; c += 256) {
    float v = base[(size_t)c * NN];
    out[(size_t)b * IMG + (size_t)c * NN + n] = (v - mean) * inv * g[c] + bta[c];
  }
}

// ---------------- asri batch-attention (tiny, [H,8,8]) -> merged o & skip ------
__global__ void asri_attn_kernel(const float* __restrict__ qkv,
                                 float* __restrict__ mo, float* __restrict__ msk) {
  int t = threadIdx.x;
  if (t >= 64) return;
  int h = t >> 3, b = t & 7;
  float sc[8];
#pragma unroll
  for (int d = 0; d < 8; d++) sc[d] = 0.f;
  for (int c = 0; c < 64; c++) {
    float qv = qkv[b * 2048 + c * 8 + h];
#pragma unroll
    for (int d = 0; d < 8; d++) sc[d] += qv * qkv[d * 2048 + 512 + c * 8 + h];
  }
  float mx = -3.0e38f;
#pragma unroll
  for (int d = 0; d < 8; d++) { sc[d] *= 0.125f; mx = fmaxf(mx, sc[d]); }
  float sum = 0.f;
#pragma unroll
  for (int d = 0; d < 8; d++) { sc[d] = __expf(sc[d] - mx); sum += sc[d]; }
  float inv = 1.f / sum;
#pragma unroll
  for (int d = 0; d < 8; d++) sc[d] *= inv;
  for (int c = 0; c < 64; c++) {
    float acc = 0.f;
#pragma unroll
    for (int d = 0; d < 8; d++) acc += sc[d] * qkv[d * 2048 + 1024 + c * 8 + h];
    mo[b * 512 + c * 8 + h]  = acc;
    msk[b * 512 + c * 8 + h] = qkv[b * 2048 + 1536 + c * 8 + h];
  }
}

// ---------------- sscm batch-attention probabilities ----------------
__global__ void sscm_attn_kernel(const float* __restrict__ qb_, const float* __restrict__ kb_,
                                 float* __restrict__ attn) {
  int t = threadIdx.x;
  if (t >= 64) return;
  int h = t >> 3, b = t & 7;
  float sc[8];
#pragma unroll
  for (int d = 0; d < 8; d++) sc[d] = 0.f;
  for (int c = 0; c < 64; c++) {
    float qv = qb_[b * 512 + c * 8 + h];
#pragma unroll
    for (int d = 0; d < 8; d++) sc[d] += qv * kb_[d * 512 + c * 8 + h];
  }
  float mx = -3.0e38f;
#pragma unroll
  for (int d = 0; d < 8; d++) { sc[d] *= 0.125f; mx = fmaxf(mx, sc[d]); }
  float sum = 0.f;
#pragma unroll
  for (int d = 0; d < 8; d++) { sc[d] = __expf(sc[d] - mx); sum += sc[d]; }
  float inv = 1.f / sum;
#pragma unroll
  for (int d = 0; d < 8; d++) attn[h * 64 + b * 8 + d] = sc[d] * inv;
}

// ---------------- sscm mix: out[b,e,s] = sum_d attn[e%8,b,d]*v[d,e,s] ----------
__global__ void sscm_mix_kernel(const float* __restrict__ attn, const float* __restrict__ v,
                                float* __restrict__ out) {
  int idx = blockIdx.x * blockDim.x + threadIdx.x;
  if (idx >= BIGF) return;
  int b = idx >> 17;
  int r = idx & (IMG - 1);
  int e = r >> 8;
  int h = e & 7;
  float acc = 0.f;
#pragma unroll
  for (int d = 0; d < 8; d++) acc += attn[h * 64 + b * 8 + d] * v[(size_t)d * IMG + r];
  out[idx] = acc;
}

// ---------------- bilinear upsample 4x4 -> 16x16 (align_corners) --------------
__global__ void upsample_kernel(const float* __restrict__ s, float* __restrict__ out) {
  int idx = blockIdx.x * blockDim.x + threadIdx.x;
  if (idx >= BIGF) return;
  int b = idx >> 17;
  int rem = idx & (IMG - 1);
  int c = rem >> 8;
  int pix = rem & 255;
  int i = pix >> 4, j = pix & 15;
  float pi = (float)(i * 3) / 15.0f;
  float pj = (float)(j * 3) / 15.0f;
  int li = (int)pi, lj = (int)pj;
  int hi = li + 1 > 3 ? 3 : li + 1;
  int hj = lj + 1 > 3 ? 3 : lj + 1;
  float wi = pi - (float)li, wj = pj - (float)lj;
  const float* sb = s + (size_t)b * 8192 + c * 16;
  float v00 = sb[li * 4 + lj], v01 = sb[li * 4 + hj];
  float v10 = sb[hi * 4 + lj], v11 = sb[hi * 4 + hj];
  out[idx] = v00 * (1.f - wi) * (1.f - wj) + v01 * (1.f - wi) * wj +
             v10 * wi * (1.f - wj) + v11 * wi * wj;
}

// ---------------- conv1x1 via f32 WMMA ----------------------------------------
// out[b,o,s] = epilogue( sum_c W[o,c] * X[b,c,s] + bias[o] )
// epilogue: optional BN scale/shift (per-o), optional ReLU, optional residual add.
__global__ __launch_bounds__(32) void conv1x1_wmma_kernel(
    const float* __restrict__ X, const float* __restrict__ W,
    const float* __restrict__ bias, const float* __restrict__ scale,
    const float* __restrict__ shift, float bn_mul,
    const float* __restrict__ residual, float* __restrict__ out, int relu) {
  const int lane = threadIdx.x;
  const int hf = lane >> 4;      // lane half (K-split)
  const int lid = lane & 15;
  const int ntile = blockIdx.x;  // 0..15 spatial tiles
  const int otile = blockIdx.y;  // 0..31 out-channel tiles
  const int b = blockIdx.z;      // 0..7
  const int n = ntile * 16 + lid;
  const int arow = otile * 16 + lid;
  const float* Xb = X + (size_t)b * IMG;
  const float* Wr = W + (size_t)arow * CC;
  v8f acc = {};
  for (int k0 = 0; k0 < CC; k0 += 4) {
    int ka = k0 + hf * 2;
    v2f a, bv;
    a.x = Wr[ka];
    a.y = Wr[ka + 1];
    bv.x = Xb[ka * NN + n];
    bv.y = Xb[(ka + 1) * NN + n];
    acc = wmma4(a, bv, acc);
  }
  const int obase = b * IMG;
#pragma unroll
  for (int v = 0; v < 8; v++) {
    int m = otile * 16 + hf * 8 + v;
    float val = acc[v] + (bias ? bias[m] : 0.f);
    if (scale) val = val * scale[m] * bn_mul + shift[m];
    if (relu) val = fmaxf(val, 0.f);
    int idx = obase + m * NN + n;
    if (residual) val += residual[idx];
    out[idx] = val;
  }
}

// ---------------- fused pixel cross-batch attention (WMMA) --------------------
// Per (h, b, 16-query-row tile): S = Q K^T (per key-batch chunk), per-row softmax
// over each 256-col chunk, O += P V.  Layout of Q/K/V: [B, c*8+h, n] (conv out).
__global__ __launch_bounds__(32) void pixel_attn_kernel(
    const float* __restrict__ Qc, const float* __restrict__ Kc,
    const float* __restrict__ Vc, float* __restrict__ O) {
  __shared__ float S[16 * 257];
  const int lane = threadIdx.x;
  const int hf = lane >> 4;
  const int lid = lane & 15;
  const int ntile = blockIdx.x;  // 0..15
  const int b = blockIdx.y;      // query batch
  const int h = blockIdx.z;      // head
  const int nq = ntile * 16 + lid;

  // Preload Q A-fragments for full K=64 (16 steps of K=4)
  v2f qa[16];
  const float* Qb = Qc + (size_t)b * IMG;
#pragma unroll
  for (int ks = 0; ks < 16; ks++) {
    int c0 = ks * 4 + hf * 2;
    qa[ks].x = Qb[(c0 * 8 + h) * NN + nq];
    qa[ks].y = Qb[((c0 + 1) * 8 + h) * NN + nq];
  }
  v8f oacc[4];
#pragma unroll
  for (int ct = 0; ct < 4; ct++) oacc[ct] = (v8f){};

  for (int qb = 0; qb < 8; qb++) {
    const float* Kb = Kc + (size_t)qb * IMG;
    const float* Vb = Vc + (size_t)qb * IMG;
    // ---- S tile: 16 queries x 256 keys of this key-batch ----
    for (int mt = 0; mt < 16; mt++) {
      v8f sacc = {};
      int mpix = mt * 16 + lid;
#pragma unroll
      for (int ks = 0; ks < 16; ks++) {
        int c0 = ks * 4 + hf * 2;
        v2f kb;
        kb.x = Kb[(c0 * 8 + h) * NN + mpix];
        kb.y = Kb[((c0 + 1) * 8 + h) * NN + mpix];
        sacc = wmma4(qa[ks], kb, sacc);
      }
#pragma unroll
      for (int v = 0; v < 8; v++) {
        int row = hf * 8 + v;
        S[row * 257 + mt * 16 + lid] = sacc[v];
      }
    }
    __syncthreads();
    // ---- softmax per row over the 256 columns of this chunk ----
    for (int r = 0; r < 16; r++) {
      float vals[8];
      float mx = -3.0e38f;
#pragma unroll
      for (int i = 0; i < 8; i++) {
        vals[i] = S[r * 257 + lane * 8 + i];
        mx = fmaxf(mx, vals[i]);
      }
      mx = warp_max(mx);
      float sum = 0.f;
#pragma unroll
      for (int i = 0; i < 8; i++) { vals[i] = __expf(vals[i] - mx); sum += vals[i]; }
      sum = warp_sum(sum);
      float inv = 1.f / sum;
#pragma unroll
      for (int i = 0; i < 8; i++) S[r * 257 + lane * 8 + i] = vals[i] * inv;
    }
    __syncthreads();
    // ---- O += P(16x256) * V(256x64) ----
#pragma unroll
    for (int ct = 0; ct < 4; ct++) {
      int c = ct * 16 + lid;
      const float* Vrow = Vb + (c * 8 + h) * NN;
      for (int ks = 0; ks < 64; ks++) {
        int k0 = ks * 4 + hf * 2;
        v2f pa, vb;
        pa.x = S[lid * 257 + k0];
        pa.y = S[lid * 257 + k0 + 1];
        vb.x = Vrow[k0];
        vb.y = Vrow[k0 + 1];
        oacc[ct] = wmma4(pa, vb, oacc[ct]);
      }
    }
    __syncthreads();
  }
  // ---- write O: [b, c*8+h, n] ----
#pragma unroll
  for (int ct = 0; ct < 4; ct++) {
#pragma unroll
    for (int v = 0; v < 8; v++) {
      int c = ct * 16 + lid;
      int npix = ntile * 16 + hf * 8 + v;
      O[(size_t)b * IMG + (c * 8 + h) * NN + npix] = oacc[ct][v];
    }
  }
}

// =============================== host side ====================================
extern "C" void kernel_launch(void* const* d_in, const int* in_sizes, int n_in,
                              void* d_out, int out_size, void* d_ws, size_t ws_size,
                              hipStream_t stream) {
  (void)in_sizes; (void)n_in; (void)out_size; (void)ws_size;
  const float* xs = (const float*)d_in[0];
  const float* xq = (const float*)d_in[1];
  const float* sa_qkv_w = (const float*)d_in[2];
  const float* sa_qkv_b = (const float*)d_in[3];
  const float* sa_pa_w  = (const float*)d_in[4];
  const float* sa_pa_b  = (const float*)d_in[5];
  const float* sa_ln1_g = (const float*)d_in[6];
  const float* sa_ln1_b = (const float*)d_in[7];
  const float* sa_f1_w  = (const float*)d_in[8];
  const float* sa_f1_b  = (const float*)d_in[9];
  const float* sa_f2_w  = (const float*)d_in[10];
  const float* sa_f2_b  = (const float*)d_in[11];
  const float* sa_ln2_g = (const float*)d_in[12];
  const float* sa_ln2_b = (const float*)d_in[13];
  const float* sa_c1_w  = (const float*)d_in[14];
  const float* sa_c1_b  = (const float*)d_in[15];
  const float* sa_c2_w  = (const float*)d_in[16];
  const float* sa_c2_b  = (const float*)d_in[17];
  const float* sa_bn_g  = (const float*)d_in[18];
  const float* sa_bn_b  = (const float*)d_in[19];
  const float* ca_q_w   = (const float*)d_in[20];
  const float* ca_q_b   = (const float*)d_in[21];
  const float* ca_k_w   = (const float*)d_in[22];
  const float* ca_k_b   = (const float*)d_in[23];
  const float* ca_v_w   = (const float*)d_in[24];
  const float* ca_v_b   = (const float*)d_in[25];
  const float* ca_pa_w  = (const float*)d_in[26];
  const float* ca_pa_b  = (const float*)d_in[27];
  const float* p_q_w    = (const float*)d_in[28];
  const float* p_q_b    = (const float*)d_in[29];
  const float* p_k_w    = (const float*)d_in[30];
  const float* p_k_b    = (const float*)d_in[31];
  const float* p_v_w    = (const float*)d_in[32];
  const float* p_v_b    = (const float*)d_in[33];
  const float* p_pa_w   = (const float*)d_in[34];
  const float* p_pa_b   = (const float*)d_in[35];
  const float* p_ln_g   = (const float*)d_in[36];
  const float* p_ln_b   = (const float*)d_in[37];
  const float* p_lin_w  = (const float*)d_in[38];
  const float* p_lin_b  = (const float*)d_in[39];
  float* out0 = (float*)d_out;
  float* out1 = out0 + BIGF;

  float* ws = (float*)d_ws;
  float* sa_s = ws;
  float* sa_q = ws + 1 * BIGF;
  float* ca_s = ws + 2 * BIGF;
  float* ca_q = ws + 3 * BIGF;
  float* T0   = ws + 4 * BIGF;
  float* T1   = ws + 5 * BIGF;
  float* T2   = ws + 6 * BIGF;
  float* T3   = ws + 7 * BIGF;
  float* sm = ws + 8 * BIGF;  // small scratch area
  float* gapA   = sm;             // 4096
  float* qkvbuf = sm + 4096;      // 16384
  float* mo     = sm + 20480;     // 4096
  float* msk    = sm + 24576;     // 4096
  float* abuf   = sm + 28672;     // 4096
  float* alnbuf = sm + 32768;     // 4096
  float* hbuf   = sm + 36864;     // 4096
  float* a2buf  = sm + 40960;     // 4096
  float* a2ln   = sm + 45056;     // 4096
  float* sbig   = sm + 49152;     // 65536
  float* gap_ss = sm + 114688;    // 4096
  float* gap_qq = sm + 118784;    // 4096
  float* qA     = sm + 122880;    // 4096
  float* kA     = sm + 126976;    // 4096
  float* qB     = sm + 131072;    // 4096
  float* kB     = sm + 135168;    // 4096
  float* attn8  = sm + 139264;    // 512

  const float bn_mul = 0.99999499993750052f;  // 1/sqrt(1+1e-5)
  dim3 cgrid(16, 32, 8);

  auto conv = [&](const float* X, const float* W, const float* bias,
                  const float* scale, const float* shift, const float* res,
                  float* out, int relu) {
    conv1x1_wmma_kernel<<<cgrid, 32, 0, stream>>>(X, W, bias, scale, shift, bn_mul,
                                                  res, out, relu);
  };

  auto asri = [&](const float* x, float* sa_out) {
    row_mean_kernel<<<4096, 32, 0, stream>>>(x, gapA, 256);
    small_gemm_kernel<<<64, 256, 0, stream>>>(gapA, sa_qkv_w, sa_qkv_b, nullptr,
                                              qkvbuf, 8, 2048, 512, 0);
    asri_attn_kernel<<<1, 64, 0, stream>>>(qkvbuf, mo, msk);
    small_gemm_kernel<<<16, 256, 0, stream>>>(mo, sa_pa_w, sa_pa_b, msk,
                                              abuf, 8, 512, 512, 0);
    ln_row_kernel<<<8, 256, 0, stream>>>(abuf, sa_ln1_g, sa_ln1_b, alnbuf, 512);
    small_gemm_kernel<<<16, 256, 0, stream>>>(alnbuf, sa_f1_w, sa_f1_b, nullptr,
                                              hbuf, 8, 512, 512, 1);
    small_gemm_kernel<<<16, 256, 0, stream>>>(hbuf, sa_f2_w, sa_f2_b, abuf,
                                              a2buf, 8, 512, 512, 0);
    ln_row_kernel<<<8, 256, 0, stream>>>(a2buf, sa_ln2_g, sa_ln2_b, a2ln, 512);
    small_gemm_kernel<<<256, 256, 0, stream>>>(a2ln, sa_c1_w, sa_c1_b, nullptr,
                                               sbig, 8, 8192, 512, 0);
    upsample_kernel<<<BIGF / 256, 256, 0, stream>>>(sbig, T0);
    // sa_out = x + relu( BN(conv1x1(up) + bias) )
    conv(T0, sa_c2_w, sa_c2_b, sa_bn_g, sa_bn_b, x, sa_out, 1);
  };

  auto sscm = [&](const float* qsm, const float* ksm, const float* x2, float* ca_out) {
    conv(x2, ca_v_w, ca_v_b, nullptr, nullptr, nullptr, T0, 0);
    sscm_attn_kernel<<<1, 64, 0, stream>>>(qsm, ksm, attn8);
    sscm_mix_kernel<<<BIGF / 256, 256, 0, stream>>>(attn8, T0, T1);
    conv(T1, ca_pa_w, ca_pa_b, nullptr, nullptr, nullptr, ca_out, 1);
  };

  auto pixel = [&](const float* x1, const float* x2, float* outp) {
    conv(x1, p_q_w, p_q_b, nullptr, nullptr, nullptr, T0, 0);
    conv(x2, p_k_w, p_k_b, nullptr, nullptr, nullptr, T1, 0);
    conv(x2, p_v_w, p_v_b, nullptr, nullptr, nullptr, T2, 0);
    pixel_attn_kernel<<<dim3(16, 8, 8), 32, 0, stream>>>(T0, T1, T2, T3);
    conv(T3, p_pa_w, p_pa_b, nullptr, nullptr, x1, T0, 0);   // + residual x1
    ln_chan_kernel<<<2048, 256, 0, stream>>>(T0, p_ln_g, p_ln_b, T1);
    conv(T1, p_lin_w, p_lin_b, nullptr, nullptr, nullptr, outp, 1);
  };

  // ---- full HRLM pipeline ----
  asri(xs, sa_s);
  asri(xq, sa_q);

  row_mean_kernel<<<4096, 32, 0, stream>>>(sa_s, gap_ss, 256);
  row_mean_kernel<<<4096, 32, 0, stream>>>(sa_q, gap_qq, 256);
  small_gemm_kernel<<<16, 256, 0, stream>>>(gap_ss, ca_q_w, ca_q_b, nullptr, qA, 8, 512, 512, 0);
  small_gemm_kernel<<<16, 256, 0, stream>>>(gap_qq, ca_k_w, ca_k_b, nullptr, kA, 8, 512, 512, 0);
  small_gemm_kernel<<<16, 256, 0, stream>>>(gap_qq, ca_q_w, ca_q_b, nullptr, qB, 8, 512, 512, 0);
  small_gemm_kernel<<<16, 256, 0, stream>>>(gap_ss, ca_k_w, ca_k_b, nullptr, kB, 8, 512, 512, 0);

  sscm(qA, kA, sa_q, ca_s);   // _sscm(sa_s, sa_q)
  sscm(qB, kB, sa_s, ca_q);   // _sscm(sa_q, sa_s)

  pixel(sa_s, ca_q, out0);    // _pixel(sa_s, ca_q)
  pixel(sa_q, ca_s, out1);    // _pixel(sa_q, ca_s)
}